// CustomPatchEmbedding_73332271612542
// MI455X (gfx1250) — compile-verified
//
#include <hip/hip_runtime.h>
#include <hip/hip_bf16.h>

// CDNA5 (gfx1250) patch-embedding: async gather to LDS + fp32 WMMA GEMM.
// Out(18432x768) = Patches(18432x768) * W^T(768x768) + bias, exact fp32 MACs
// on the matrix pipe via V_WMMA_F32_16X16X4_F32.
//
// Round 4: gather now uses GLOBAL_LOAD_ASYNC_TO_LDS_B32 (ASYNCcnt-tracked,
// no VGPR staging, no ds_store) + s_wait_asynccnt before the barrier.
// Compute: M-block = 48 patches (144.8 KB LDS), each weight (B) fragment
// from L2 feeds THREE WMMAs (L2 weight traffic 0.9 GB), three independent
// accumulator chains for pipeline overlap.

typedef __attribute__((ext_vector_type(2))) float v2f;
typedef __attribute__((ext_vector_type(8))) float v8f;

#define PH      16
#define PW      16
#define CCH     3
#define HH      384
#define WW      384
#define NPATCH  576
#define BATCH   32
#define EMBED   768
#define KDIM    768          // in_feat = 3*16*16
#define LDS_STRIDE 772       // 768 + 4 pad: stride%64==4 -> conflict-free A reads
#define MBLK    48           // patches per workgroup (three 16-row WMMA subtiles)

__global__ __launch_bounds__(256)
void patch_embed_wmma_f32(const float* __restrict__ x,
                          const int*   __restrict__ h_idx,
                          const int*   __restrict__ w_idx,
                          const float* __restrict__ proj_w,
                          const float* __restrict__ proj_b,
                          float*       __restrict__ out)
{
    __shared__ float sP[MBLK * LDS_STRIDE];  // 48 patches x 768 features (+pad) ~144.8 KB

    const int tile = blockIdx.x;             // 0..383, 48 patches per tile
    const int t    = threadIdx.x;            // 0..255

    // ------------- gather: 48 patches -> LDS via async copies ---------------
    // phase thread map: patch m = phase*16 + t/16, pixel column pw = t%16.
    // feature index f = c*256 + ph*16 + pw  (matches reference transpose order)
    // GLOBAL_LOAD_ASYNC_TO_LDS_B32: per-lane global->LDS, tracked by ASYNCcnt.
    #pragma unroll
    for (int phase = 0; phase < 3; ++phase) {
        const int m  = (t >> 4) + 16 * phase;
        const int pw = t & 15;
        const int p  = tile * MBLK + m;      // global patch id
        const int b  = p / NPATCH;
        const int n  = p - b * NPATCH;
        const int h0 = h_idx[b * NPATCH + n];
        const int w0 = w_idx[b * NPATCH + n];
        const float* __restrict__ xb = x + (size_t)b * (CCH * HH * WW) + (size_t)w0 + pw;
        float* __restrict__ dst = sP + m * LDS_STRIDE + pw;
        #pragma unroll
        for (int j = 0; j < 48; ++j) {       // j = c*16 + ph
            const int c  = j >> 4;
            const int ph = j & 15;
            // generic LDS pointer: low 32 bits == LDS byte address (ISA 10.2)
            const unsigned           ldsa = (unsigned)(size_t)(dst + j * 16);
            const unsigned long long ga   =
                (unsigned long long)(size_t)(xb + c * (HH * WW) + (h0 + ph) * WW);
            asm volatile("global_load_async_to_lds_b32 %0, %1, off"
                         :: "v"(ldsa), "v"(ga)
                         : "memory");
        }
    }
    asm volatile("s_wait_asynccnt 0x0" ::: "memory");   // LDS fill complete (this wave)
    __syncthreads();                                     // visible to all waves

    // ---------------- compute: 6 embed tiles per wave, 3 M-subtiles each ---
    const int wave = t >> 5;
    const int lane = t & 31;
    const int l16  = lane & 15;
    const int hl   = lane >> 4;              // 0 -> K+{0,1}, 1 -> K+{2,3}

    // A fragment bases: lane holds rows M = l16, l16+16, l16+32
    const float* __restrict__ aBase0 = sP + l16 * LDS_STRIDE + 2 * hl;
    const float* __restrict__ aBase1 = aBase0 + 16 * LDS_STRIDE;
    const float* __restrict__ aBase2 = aBase0 + 32 * LDS_STRIDE;
    const size_t outRow0 = (size_t)tile * MBLK;

    for (int q = 0; q < 6; ++q) {
        const int e0 = (wave + 8 * q) * 16;  // embed-tile origin
        // B fragment: lane holds column N=l16 of W^T, i.e. row e0+l16 of W
        const float* __restrict__ bBase = proj_w + (size_t)(e0 + l16) * KDIM + 2 * hl;

        v8f acc0 = {};
        v8f acc1 = {};
        v8f acc2 = {};
        #pragma unroll 8
        for (int k0 = 0; k0 < KDIM; k0 += 4) {
            v2f bm = *(const v2f*)(bBase + k0);    // one weight fetch ...
            v2f a0 = *(const v2f*)(aBase0 + k0);
            v2f a1 = *(const v2f*)(aBase1 + k0);
            v2f a2 = *(const v2f*)(aBase2 + k0);
            // ... feeds three independent fp32 WMMAs (D = A(16x4)*B(4x16)+C, RNE)
            acc0 = __builtin_amdgcn_wmma_f32_16x16x4_f32(
                       false, a0, false, bm, (short)0, acc0, false, false);
            acc1 = __builtin_amdgcn_wmma_f32_16x16x4_f32(
                       false, a1, false, bm, (short)0, acc1, false, false);
            acc2 = __builtin_amdgcn_wmma_f32_16x16x4_f32(
                       false, a2, false, bm, (short)0, acc2, false, false);
        }

        const float bias = proj_b[e0 + l16];       // N = l16 for this lane
        // D layout: VGPR i -> row (i + 8*hl), col l16
        float* __restrict__ orow0 = out + (outRow0 + 8 * hl) * EMBED + e0 + l16;
        float* __restrict__ orow1 = orow0 + (size_t)16 * EMBED;
        float* __restrict__ orow2 = orow0 + (size_t)32 * EMBED;
        #pragma unroll
        for (int i = 0; i < 8; ++i) {
            orow0[(size_t)i * EMBED] = acc0[i] + bias;
            orow1[(size_t)i * EMBED] = acc1[i] + bias;
            orow2[(size_t)i * EMBED] = acc2[i] + bias;
        }
    }
}

extern "C" void kernel_launch(void* const* d_in, const int* in_sizes, int n_in,
                              void* d_out, int out_size, void* d_ws, size_t ws_size,
                              hipStream_t stream) {
    (void)in_sizes; (void)n_in; (void)out_size; (void)d_ws; (void)ws_size;
    const float* x      = (const float*)d_in[0];
    const int*   h_idx  = (const int*)  d_in[1];
    const int*   w_idx  = (const int*)  d_in[2];
    const float* proj_w = (const float*)d_in[3];
    const float* proj_b = (const float*)d_in[4];
    float*       out    = (float*)d_out;

    const int m_tiles = (BATCH * NPATCH) / MBLK;   // 384
    hipLaunchKernelGGL(patch_embed_wmma_f32,
                       dim3(m_tiles), dim3(256), 0, stream,
                       x, h_idx, w_idx, proj_w, proj_b, out);
}